// HPWL_60043642798465
// MI455X (gfx1250) — compile-verified
//
#include <hip/hip_runtime.h>
#include <stdint.h>

// ---------------------------------------------------------------------------
// HPWL on MI455X (gfx1250): bandwidth-bound segmented min/max + weighted sum.
//   pos[4M,2] f32, flat_netpin[4M] i64, pin2net[4M] i64 (sorted),
//   net_weights[1M] f32, net_mask[1M] bool  ->  out[2] f32
// ~135 MB of traffic => ~6us roofline at 23.3 TB/s. Strategy:
//   * monotone float<->uint map so the segmented reduce uses integer
//     atomicMin/Max (order-independent => deterministic, no fences)
//   * sortedness of pin2net => merge consecutive same-net pins per thread
//     before issuing atomics (avg 4 pins/net)
//   * streamed inputs loaded as b128 vectors with non-temporal hint (TH=NT)
//     so the one-shot 96 MB stream does not evict the atomic workspace in L2
//   * final weighted sum via V_WMMA_F32_16X16X4_F32: A = ones(16x4),
//     B = 64 weighted-bbox values per wave-iteration, C accumulates.
// ---------------------------------------------------------------------------

typedef float     v2f  __attribute__((ext_vector_type(2)));
typedef float     v8f  __attribute__((ext_vector_type(8)));
typedef long long v2i64 __attribute__((ext_vector_type(2)));

static constexpr int G_BLOCKS = 512;   // fixed grid for k_net (deterministic partials)
static constexpr int BLOCK    = 256;   // 8 waves per block (wave32)
static constexpr int PPT      = 8;     // pins per thread in scatter kernel

// Monotone mapping: float order == unsigned order.
__device__ __forceinline__ unsigned int f2ord(float f) {
  unsigned int u = __float_as_uint(f);
  return (u & 0x80000000u) ? ~u : (u | 0x80000000u);
}
__device__ __forceinline__ float ord2f(unsigned int v) {
  return (v & 0x80000000u) ? __uint_as_float(v ^ 0x80000000u)
                           : __uint_as_float(~v);
}

// mm[n] = { min_x, max_x, min_y, max_y } as ordered uints.
__global__ void k_init(uint4* __restrict__ mm, int N) {
  int n = blockIdx.x * blockDim.x + threadIdx.x;
  if (n < N) mm[n] = make_uint4(0xFFFFFFFFu, 0u, 0xFFFFFFFFu, 0u);
}

__device__ __forceinline__ void flushmm(unsigned int* __restrict__ mm, long long n,
                                        unsigned int mnx, unsigned int mxx,
                                        unsigned int mny, unsigned int mxy) {
  unsigned int* p = mm + 4ll * n;
  atomicMin(p + 0, mnx);
  atomicMax(p + 1, mxx);
  atomicMin(p + 2, mny);
  atomicMax(p + 3, mxy);
}

// Each thread owns PPT consecutive pins; pin2net is sorted, so runs of equal
// net ids are merged in registers before the (rare) atomic flush.
__global__ void k_scatter(const float* __restrict__ pos,
                          const long long* __restrict__ flat,
                          const long long* __restrict__ p2n,
                          unsigned int* __restrict__ mm, int P) {
  long long base = (long long)(blockIdx.x * blockDim.x + threadIdx.x) * PPT;
  if (base >= P) return;

  long long cur = -1;
  unsigned int mnx = 0, mxx = 0, mny = 0, mxy = 0;

  if (base + PPT <= P) {
    // Fast path: 64B-aligned contiguous chunks -> 4x b128 NT loads per array.
    long long nets[PPT], fps[PPT];
    #pragma unroll
    for (int k = 0; k < PPT / 2; ++k) {
      v2i64 nv = __builtin_nontemporal_load((const v2i64*)(p2n + base) + k);
      v2i64 fv = __builtin_nontemporal_load((const v2i64*)(flat + base) + k);
      nets[2 * k] = nv[0]; nets[2 * k + 1] = nv[1];
      fps [2 * k] = fv[0]; fps [2 * k + 1] = fv[1];
    }
    #pragma unroll
    for (int j = 0; j < PPT; ++j) {
      v2f c = __builtin_nontemporal_load((const v2f*)pos + fps[j]);  // b64 gather
      unsigned int ux = f2ord(c[0]), uy = f2ord(c[1]);
      long long n = nets[j];
      if (n != cur) {
        if (cur >= 0) flushmm(mm, cur, mnx, mxx, mny, mxy);
        cur = n; mnx = ux; mxx = ux; mny = uy; mxy = uy;
      } else {
        mnx = min(mnx, ux); mxx = max(mxx, ux);
        mny = min(mny, uy); mxy = max(mxy, uy);
      }
    }
  } else {
    // Tail path (only the final partial chunk).
    int cnt = (int)(P - base);
    for (int j = 0; j < cnt; ++j) {
      long long i  = base + j;
      long long n  = p2n[i];
      long long fp = flat[i];
      unsigned int ux = f2ord(pos[2 * fp + 0]);
      unsigned int uy = f2ord(pos[2 * fp + 1]);
      if (n != cur) {
        if (cur >= 0) flushmm(mm, cur, mnx, mxx, mny, mxy);
        cur = n; mnx = ux; mxx = ux; mny = uy; mxy = uy;
      } else {
        mnx = min(mnx, ux); mxx = max(mxx, ux);
        mny = min(mny, uy); mxy = max(mxy, uy);
      }
    }
  }
  flushmm(mm, cur, mnx, mxx, mny, mxy);
}

// Weighted bbox sum via WMMA accumulation. Each wave handles 64-net tiles:
// lane supplies nets (t*64+lane) and (t*64+lane+32) as the two B VGPRs.
// D[m][n] = sum_k 1 * B[k][n] + C[m][n]  => columns accumulate all values;
// rows of D are duplicates, so sum over 32 lanes of D-vgpr0 == 2 * total.
__global__ void k_net(const uint4* __restrict__ mm,
                      const float* __restrict__ w,
                      const unsigned char* __restrict__ mask,
                      float2* __restrict__ partials, int N) {
  const int lane        = threadIdx.x & 31;
  const int waveInBlk   = threadIdx.x >> 5;
  const int wavesPerBlk = BLOCK / 32;
  const int gwave       = blockIdx.x * wavesPerBlk + waveInBlk;
  const int nwaves      = G_BLOCKS * wavesPerBlk;
  const int T           = (N + 63) / 64;

  v2f ones; ones[0] = 1.0f; ones[1] = 1.0f;
  v8f ax = {}; v8f ay = {};

  for (int t = gwave; t < T; t += nwaves) {   // uniform per wave: EXEC all-1s at WMMA
    int n0 = t * 64 + lane;
    int n1 = n0 + 32;
    float x0 = 0.f, y0 = 0.f, x1 = 0.f, y1 = 0.f;
    if (n0 < N) {
      uint4 m = mm[n0];
      float ww = w[n0] * (mask[n0] ? 1.0f : 0.0f);
      float bx = (m.y > m.x) ? (ord2f(m.y) - ord2f(m.x)) : 0.0f;  // empty net => max<min => 0
      float by = (m.w > m.z) ? (ord2f(m.w) - ord2f(m.z)) : 0.0f;
      x0 = ww * bx; y0 = ww * by;
    }
    if (n1 < N) {
      uint4 m = mm[n1];
      float ww = w[n1] * (mask[n1] ? 1.0f : 0.0f);
      float bx = (m.y > m.x) ? (ord2f(m.y) - ord2f(m.x)) : 0.0f;
      float by = (m.w > m.z) ? (ord2f(m.w) - ord2f(m.z)) : 0.0f;
      x1 = ww * bx; y1 = ww * by;
    }
    v2f bx; bx[0] = x0; bx[1] = x1;
    v2f by; by[0] = y0; by[1] = y1;
    ax = __builtin_amdgcn_wmma_f32_16x16x4_f32(false, ones, false, bx,
                                               (short)0, ax, false, false);
    ay = __builtin_amdgcn_wmma_f32_16x16x4_f32(false, ones, false, by,
                                               (short)0, ay, false, false);
  }

  float sx = ax[0];
  float sy = ay[0];
  #pragma unroll
  for (int off = 16; off > 0; off >>= 1) {
    sx += __shfl_down(sx, off, 32);
    sy += __shfl_down(sy, off, 32);
  }

  __shared__ float2 red[BLOCK / 32];
  if (lane == 0) red[waveInBlk] = make_float2(sx * 0.5f, sy * 0.5f);
  __syncthreads();
  if (threadIdx.x == 0) {
    float px = 0.f, py = 0.f;
    for (int i = 0; i < wavesPerBlk; ++i) { px += red[i].x; py += red[i].y; }
    partials[blockIdx.x] = make_float2(px, py);
  }
}

// Fixed-order combine of the 512 block partials => deterministic output.
__global__ void k_final(const float2* __restrict__ partials, float* __restrict__ out) {
  if (threadIdx.x == 0 && blockIdx.x == 0) {
    float sx = 0.f, sy = 0.f;
    for (int i = 0; i < G_BLOCKS; ++i) { sx += partials[i].x; sy += partials[i].y; }
    out[0] = sx;
    out[1] = sy;
  }
}

extern "C" void kernel_launch(void* const* d_in, const int* in_sizes, int n_in,
                              void* d_out, int out_size, void* d_ws, size_t ws_size,
                              hipStream_t stream) {
  const float*         pos  = (const float*)d_in[0];
  const long long*     flat = (const long long*)d_in[1];
  const long long*     p2n  = (const long long*)d_in[2];
  const float*         w    = (const float*)d_in[3];
  const unsigned char* mask = (const unsigned char*)d_in[4];

  const int P = in_sizes[0] / 2;  // number of pins
  const int N = in_sizes[3];      // number of nets

  // Workspace layout: [0, 16*N) per-net ordered min/max; then block partials.
  uint4*  mm       = (uint4*)d_ws;
  float2* partials = (float2*)((char*)d_ws + (size_t)N * sizeof(uint4));

  k_init<<<(N + BLOCK - 1) / BLOCK, BLOCK, 0, stream>>>(mm, N);

  int scatterThreads = (P + PPT - 1) / PPT;
  k_scatter<<<(scatterThreads + BLOCK - 1) / BLOCK, BLOCK, 0, stream>>>(
      pos, flat, p2n, (unsigned int*)mm, P);

  k_net<<<G_BLOCKS, BLOCK, 0, stream>>>(mm, w, mask, partials, N);

  k_final<<<1, 32, 0, stream>>>(partials, (float*)d_out);
}